// DySample_16028817949222
// MI455X (gfx1250) — compile-verified
//
#include <hip/hip_runtime.h>
#include <math.h>

// ---------------------------------------------------------------------------
// DySample fused implementation for gfx1250 (wave32, WMMA f32 16x16x4)
//   x        : (8, 64, 128, 128) f32
//   w_offset : (32, 64)   b_offset : (32,)
//   w_scope  : (32, 64)
//   w_end    : (64, 64)   b_end    : (64,)
//   out      : (8, 64, 256, 256) f32
// ---------------------------------------------------------------------------

typedef __attribute__((ext_vector_type(2))) float v2f;
typedef __attribute__((ext_vector_type(8))) float v8f;
// 4-byte-aligned float2 for possibly-unaligned paired-tap gathers
typedef float v2fu __attribute__((ext_vector_type(2), aligned(4)));
// Parameter type required by the async-to-LDS builtins (from clang diagnostic)
typedef int vec4i __attribute__((vector_size(4 * sizeof(int))));

#define HS 128
#define WSW 128
#define CS 64
#define HW (HS * WSW)
#define OH (2 * HS)
#define OW (2 * WSW)
#define WPAD 68  // LDS weight row stride (floats): conflict-free b64 reads

#define AS1 __attribute__((address_space(1)))
#define AS3 __attribute__((address_space(3)))

#if __has_builtin(__builtin_amdgcn_global_load_async_to_lds_b128)
#define HAVE_ASYNC_LDS 1
#else
#define HAVE_ASYNC_LDS 0
#endif

#if HAVE_ASYNC_LDS
#define ASYNC_B128(gp, lp) \
  __builtin_amdgcn_global_load_async_to_lds_b128( \
      (AS1 vec4i*)(gp), (AS3 vec4i*)(lp), /*offset=*/0, /*cpol=*/0)
#endif

static __device__ __forceinline__ void wait_async_then_barrier() {
#if HAVE_ASYNC_LDS
#if __has_builtin(__builtin_amdgcn_s_wait_asynccnt)
  __builtin_amdgcn_s_wait_asynccnt(0);
#else
  asm volatile("s_wait_asynccnt 0" ::: "memory");
#endif
#endif
  __syncthreads();
}

static __device__ __forceinline__ v8f wmma_f32(v2f a, v2f b, v8f c) {
  // D = A(16x4) * B(4x16) + C(16x16), fp32
  return __builtin_amdgcn_wmma_f32_16x16x4_f32(false, a, false, b, (short)0, c,
                                               false, false);
}

static __device__ __forceinline__ float sigmoidf_(float t) {
  return 1.0f / (1.0f + __expf(-t));
}

// Stage a 64x64 f32 weight matrix (row-major, contiguous) into LDS with
// row stride WPAD. Async DMA (ASYNCcnt) when available; completion is
// enforced by wait_async_then_barrier() at the consumer.
static __device__ __forceinline__ void stage_weights(const float* __restrict__ g,
                                                     float* l, int tid) {
#if HAVE_ASYNC_LDS
#pragma unroll
  for (int q = tid; q < 1024; q += 256) {  // 1024 float4 chunks
    const int oc = q >> 4;
    const int c4 = (q & 15) * 4;
    ASYNC_B128(g + q * 4, l + oc * WPAD + c4);
  }
#else
#pragma unroll
  for (int i = tid; i < 64 * 64; i += 256) {
    l[(i >> 6) * WPAD + (i & 63)] = g[i];
  }
#endif
}

// ---------------------------------------------------------------------------
// Kernel 1: offsets -> absolute clamped sample coords, pixel-shuffled layout.
// coords[b][gi][oh][ow][2] = {gx, gy} in source-pixel units, clamped [0,127].
// grid = (H, B), block = 256 (8 waves); wave w handles pixels [16w,16w+16).
// ---------------------------------------------------------------------------
__global__ void __launch_bounds__(256)
k_coords(const float* __restrict__ x, const float* __restrict__ w_off,
         const float* __restrict__ b_off, const float* __restrict__ w_scope,
         float* __restrict__ coords) {
  // Combined weight [64 oc][64 c]: oc<32 -> w_offset (rows 0-31),
  // oc>=32 -> w_scope (rows 32-63). Both staged async back-to-back.
  __shared__ float lds_w[64 * WPAD];

  const int h = blockIdx.x;
  const int b = blockIdx.y;
  const int wave = threadIdx.x >> 5;
  const int lane = threadIdx.x & 31;
  const int l15 = lane & 15;
  const int chalf = (lane < 16) ? 0 : 2;  // K sub-pair held by this lane half
  const int wtile = wave * 16;

#if HAVE_ASYNC_LDS
  // Each 32x64 matrix = 512 float4 chunks; 256 threads -> 2 chunks each.
  {
    const int q = threadIdx.x;  // w_offset rows 0..31 -> LDS rows 0..31
    const int q2 = threadIdx.x + 256;
    ASYNC_B128(w_off + q * 4, lds_w + (q >> 4) * WPAD + (q & 15) * 4);
    ASYNC_B128(w_off + q2 * 4, lds_w + (q2 >> 4) * WPAD + (q2 & 15) * 4);
    ASYNC_B128(w_scope + q * 4, lds_w + ((q >> 4) + 32) * WPAD + (q & 15) * 4);
    ASYNC_B128(w_scope + q2 * 4,
               lds_w + ((q2 >> 4) + 32) * WPAD + (q2 & 15) * 4);
  }
#else
#pragma unroll
  for (int i = threadIdx.x; i < 64 * 64; i += 256) {
    const int oc = i >> 6;
    const int c = i & 63;
    lds_w[oc * WPAD + c] =
        (oc < 32) ? w_off[oc * 64 + c] : w_scope[(oc - 32) * 64 + c];
  }
#endif

  // Per-lane base pointer: all 32 A-loads use int immediate plane offsets.
  const float* xl = x + (size_t)b * CS * HW + h * WSW + wtile + l15;

  // Accumulators: nt0/nt1 = offset-x / offset-y channels, nt2/nt3 = scope.
  v8f acc0, acc1, acc2, acc3;
  const float bo0 = b_off[l15];
  const float bo1 = b_off[16 + l15];
#pragma unroll
  for (int i = 0; i < 8; ++i) {
    acc0[i] = bo0;
    acc1[i] = bo1;
    acc2[i] = 0.0f;
    acc3[i] = 0.0f;
  }

  wait_async_then_barrier();  // weights visible in LDS for all waves

#pragma unroll
  for (int kk = 0; kk < 16; ++kk) {
    const int c0 = 4 * kk + chalf;
    v2f a;
    a.x = xl[c0 * HW];
    a.y = xl[c0 * HW + HW];

    // B fragments from LDS (ds_load b64 pairs, bank-conflict-free via WPAD).
    const v2f b0 = *(const v2f*)&lds_w[(l15) * WPAD + c0];        // off-x
    const v2f b1 = *(const v2f*)&lds_w[(16 + l15) * WPAD + c0];   // off-y
    const v2f b2 = *(const v2f*)&lds_w[(32 + l15) * WPAD + c0];   // scope 0-15
    const v2f b3 = *(const v2f*)&lds_w[(48 + l15) * WPAD + c0];   // scope 16-31
    acc0 = wmma_f32(a, b0, acc0);
    acc1 = wmma_f32(a, b1, acc1);
    acc2 = wmma_f32(a, b2, acc2);
    acc3 = wmma_f32(a, b3, acc3);
  }

  // Epilogue: lane owns sub-position n = l15 -> (gi, r1, r2); rows = pixels.
  const int gi = l15 >> 2;
  const int r1 = (l15 >> 1) & 1;  // y sub-pixel
  const int r2 = l15 & 1;         // x sub-pixel
  const float initx = r2 ? 0.25f : -0.25f;
  const float inity = r1 ? 0.25f : -0.25f;
  const int mbase = (lane < 16) ? 0 : 8;
  const int oh = 2 * h + r1;

#pragma unroll
  for (int r = 0; r < 8; ++r) {
    const int w = wtile + mbase + r;
    const float offx = acc0[r] * sigmoidf_(acc2[r]) * 0.5f + initx;
    const float offy = acc1[r] * sigmoidf_(acc3[r]) * 0.5f + inity;
    float gx = (float)w + offx;
    float gy = (float)h + offy;
    gx = fminf(fmaxf(gx, 0.0f), (float)(WSW - 1));
    gy = fminf(fmaxf(gy, 0.0f), (float)(HS - 1));
    const int ow = 2 * w + r2;
    const size_t ci = ((((size_t)b * 4 + gi) * OH + oh) * OW + ow) * 2;
    coords[ci + 0] = gx;
    coords[ci + 1] = gy;
  }
}

// Bilinear from two row-pairs. Border handled upstream: base index clamped to
// W-2/H-2 and weight allowed to reach exactly 1.0 (selects clamped texel).
static __device__ __forceinline__ float bilin2(v2fu top, v2fu bot, float wx,
                                               float wy) {
  const float t = fmaf(wx, top.y - top.x, top.x);
  const float bo = fmaf(wx, bot.y - bot.x, bot.x);
  return fmaf(wy, bo - t, t);
}

// ---------------------------------------------------------------------------
// Kernel 2: fused bilinear grid-sample + final 1x1 conv (64->64) via WMMA.
// grid = (2, 2H, B), block = 256 (8 waves); block covers 128 consecutive ow
// of one output row; each wave owns a 16-pixel tile. Weight DMA overlaps the
// entire gather phase.
// ---------------------------------------------------------------------------
__global__ void __launch_bounds__(256)
k_sample_conv(const float* __restrict__ x, const float* __restrict__ coords,
              const float* __restrict__ w_end, const float* __restrict__ b_end,
              float* __restrict__ out) {
  __shared__ float lds_w[64 * WPAD];  // staged w_end
  __shared__ float tile[64 * 132];    // D-tile transpose buffer

  const int b = blockIdx.z;
  const int oh = blockIdx.y;
  const int owblk = blockIdx.x * 128;
  const int wave = threadIdx.x >> 5;
  const int lane = threadIdx.x & 31;
  const int l15 = lane & 15;
  const int chalf = (lane < 16) ? 0 : 2;
  const int ow = owblk + wave * 16 + l15;  // this lane's output pixel (M index)

  // Kick off async weight DMA; it completes under the gather phase.
  stage_weights(w_end, lds_w, threadIdx.x);

  const float* xb = x + (size_t)b * CS * HW;

  // ---- Gather phase: sampled values land directly in A-fragment layout ----
  v2f afr[16];
#pragma unroll
  for (int gi = 0; gi < 4; ++gi) {
    const size_t ci = ((((size_t)b * 4 + gi) * OH + oh) * OW + ow) * 2;
    const float gx = coords[ci + 0];
    const float gy = coords[ci + 1];
    // Base-clamped bilinear: weight may reach 1.0 exactly at the border.
    const int ixb = min((int)floorf(gx), WSW - 2);
    const int iyb = min((int)floorf(gy), HS - 2);
    const float wx = gx - (float)ixb;
    const float wy = gy - (float)iyb;
    // Per-lane base for this group; all 16 taps use int immediate offsets.
    const float* pg = xb + (iyb * WSW + ixb);
#pragma unroll
    for (int j = 0; j < 4; ++j) {
      const int c0 = gi * 16 + 4 * j + chalf;
      const int o0 = c0 * HW;
      const v2fu a00 = *(const v2fu*)(pg + o0);
      const v2fu a01 = *(const v2fu*)(pg + o0 + WSW);
      const v2fu b00 = *(const v2fu*)(pg + o0 + HW);
      const v2fu b01 = *(const v2fu*)(pg + o0 + HW + WSW);
      v2f a;
      a.x = bilin2(a00, a01, wx, wy);
      a.y = bilin2(b00, b01, wx, wy);
      afr[gi * 4 + j] = a;
    }
  }

  // ---- GEMM phase: (16 pix x 64 ch) * w_end^T(64x64) -> 16x64 ----
  v8f acc[4];
#pragma unroll
  for (int nt = 0; nt < 4; ++nt)
#pragma unroll
    for (int i = 0; i < 8; ++i) acc[nt][i] = 0.0f;

  wait_async_then_barrier();  // weight DMA done, LDS visible block-wide

#pragma unroll
  for (int kk = 0; kk < 16; ++kk) {
    const int c0 = 4 * kk + chalf;
#pragma unroll
    for (int nt = 0; nt < 4; ++nt) {
      const int oc = nt * 16 + l15;  // N index held by this lane
      const v2f bv = *(const v2f*)&lds_w[oc * WPAD + c0];
      acc[nt] = wmma_f32(afr[kk], bv, acc[nt]);
    }
  }

  // ---- Transpose D tiles through LDS for coalesced global stores ----
  const int pixl = wave * 16 + ((lane < 16) ? 0 : 8);
#pragma unroll
  for (int nt = 0; nt < 4; ++nt) {
    const int oc = nt * 16 + l15;
    float4 lo = make_float4(acc[nt][0], acc[nt][1], acc[nt][2], acc[nt][3]);
    float4 hi = make_float4(acc[nt][4], acc[nt][5], acc[nt][6], acc[nt][7]);
    *(float4*)&tile[oc * 132 + pixl + 0] = lo;
    *(float4*)&tile[oc * 132 + pixl + 4] = hi;
  }
  __syncthreads();

  // 64 oc x 128 pix = 2048 float4 units, coalesced, bias fused.
#pragma unroll
  for (int i = threadIdx.x; i < 64 * 32; i += 256) {
    const int oc = i >> 5;
    const int p4 = (i & 31) * 4;
    float4 v = *(const float4*)&tile[oc * 132 + p4];
    const float be = b_end[oc];
    v.x += be;
    v.y += be;
    v.z += be;
    v.w += be;
    *(float4*)&out[(((size_t)b * 64 + oc) * OH + oh) * OW + owblk + p4] = v;
  }
}

// ---------------------------------------------------------------------------

extern "C" void kernel_launch(void* const* d_in, const int* in_sizes, int n_in,
                              void* d_out, int out_size, void* d_ws,
                              size_t ws_size, hipStream_t stream) {
  const float* x = (const float*)d_in[0];
  const float* w_offset = (const float*)d_in[1];
  const float* b_offset = (const float*)d_in[2];
  const float* w_scope = (const float*)d_in[3];
  const float* w_end = (const float*)d_in[4];
  const float* b_end = (const float*)d_in[5];
  float* out = (float*)d_out;

  // coords scratch: (8, 4, 256, 256, 2) f32 = 16.8 MB
  float* coords = (float*)d_ws;

  dim3 g1(HS, 8);
  k_coords<<<g1, 256, 0, stream>>>(x, w_offset, b_offset, w_scope, coords);

  dim3 g2(2, OH, 8);
  k_sample_conv<<<g2, 256, 0, stream>>>(x, coords, w_end, b_end, out);
}